// LDM_5927054868953
// MI455X (gfx1250) — compile-verified
//
#include <hip/hip_runtime.h>

typedef __attribute__((ext_vector_type(2))) float v2f;
typedef __attribute__((ext_vector_type(8))) float v8f;

#define EPS 1e-6f

__global__ void ldm_init_out(float* out) {
    if (threadIdx.x == 0 && blockIdx.x == 0) out[0] = 0.0f;
}

// ---------------------------------------------------------------------------
// z_pdist1 (fused, negated): per 16-row block of latent_l, sweep R and U in
// 16x16 tiles. d^2 via V_WMMA_F32_16X16X4_F32 with the A block register-
// resident; B tiles double-buffered in registers so one tile's 32 loads are
// in flight while the previous tile's WMMA chain + exp/sqrt epilogue runs.
// One wave per i-block; 4 independent waves per 128-thread block.
// ---------------------------------------------------------------------------
__global__ __launch_bounds__(128)
void ldm_z1_kernel(const float* __restrict__ L, const float* __restrict__ R,
                   const float* __restrict__ U, const float* __restrict__ rho,
                   const float* __restrict__ nu, const float* __restrict__ tau,
                   int N, int S, int B, float* __restrict__ out)
{
    const int lane = threadIdx.x & 31;
    const int ib   = blockIdx.x * 4 + (threadIdx.x >> 5);
    if (ib * 16 >= N) return;                 // wave-uniform exit (EXEC stays all-1s)

    const int i0  = ib * 16;
    const int row = lane & 15;                // A row / B col / C col index
    const int h   = lane >> 4;                // K-half for A/B loads; M-half for C

    // ---- Stage A block (latent_l + EPS) in registers; per-row |a|^2 --------
    v2f   a[32];
    float asq_p = 0.0f;
    const float* Arow = L + (size_t)(i0 + row) * 128 + 2 * h;
    #pragma unroll
    for (int kb = 0; kb < 32; ++kb) {
        float2 t = *(const float2*)(Arow + kb * 4);
        float x = t.x + EPS, y = t.y + EPS;
        a[kb].x = x; a[kb].y = y;
        asq_p += x * x + y * y;
    }
    const float asq = asq_p + __shfl_xor(asq_p, 16, 32);  // lane m holds |a_{m%16}|^2

    float accj[8] = {0,0,0,0,0,0,0,0};
    float acck[8] = {0,0,0,0,0,0,0,0};

    // Pure load phase: 32 back-to-back b64 loads into a register buffer.
    auto loadT = [&](const float* __restrict__ p, v2f* buf) {
        #pragma unroll
        for (int kb = 0; kb < 32; ++kb) {
            float2 t = *(const float2*)(p + kb * 4);
            buf[kb].x = t.x; buf[kb].y = t.y;
        }
    };

    // Compute phase: |b|^2 + 32-deep WMMA chain + transcendental epilogue.
    auto computeT = [&](const v2f* buf, const float* __restrict__ bias, int j0,
                        float* acc) {
        v8f   c = {};
        float bsq_p = 0.0f;
        #pragma unroll
        for (int kb = 0; kb < 32; ++kb) {
            bsq_p += buf[kb].x * buf[kb].x + buf[kb].y * buf[kb].y;
            c = __builtin_amdgcn_wmma_f32_16x16x4_f32(
                    false, a[kb], false, buf[kb], (short)0, c, false, false);
        }
        const float bsq = bsq_p + __shfl_xor(bsq_p, 16, 32);  // |b_{lane%16}|^2
        const float bn  = bias[j0 + row];                     // nu_j / tau_k, col = lane%16
        #pragma unroll
        for (int r = 0; r < 8; ++r) {
            // C layout: VGPR r, lanes 0-15 -> M=r, lanes 16-31 -> M=r+8; N = lane%16
            float asqM = __shfl(asq, r + 8 * h, 32);
            float d2   = asqM + bsq - 2.0f * c[r];
            float dist = sqrtf(fmaxf(d2, 0.0f));
            float m    = __expf(-(dist + EPS));
            acc[r]    += __expf(bn - m);
        }
    };

    auto pass = [&](const float* __restrict__ Mat, const float* __restrict__ bias,
                    int ntiles, float* acc) {
        const float* base = Mat + (size_t)row * 128 + 2 * h;
        v2f b0[32], b1[32];
        loadT(base, b0);                                   // tile 0
        for (int jt = 0; jt < ntiles; jt += 2) {
            if (jt + 1 < ntiles) loadT(base + (size_t)(jt + 1) * 16 * 128, b1);
            computeT(b0, bias, jt * 16, acc);
            if (jt + 1 < ntiles) {
                if (jt + 2 < ntiles) loadT(base + (size_t)(jt + 2) * 16 * 128, b0);
                computeT(b1, bias, (jt + 1) * 16, acc);
            }
        }
    };

    pass(R, nu,  S / 16, accj);
    pass(U, tau, B / 16, acck);

    // Row sums: reduce each 16-lane half; lanes 0-15 -> rows r, 16-31 -> rows r+8.
    #pragma unroll
    for (int r = 0; r < 8; ++r) {
        #pragma unroll
        for (int off = 1; off < 16; off <<= 1) {
            accj[r] += __shfl_xor(accj[r], off, 32);
            acck[r] += __shfl_xor(acck[r], off, 32);
        }
    }

    if (row == 0) {  // lanes 0 and 16
        float part = 0.0f;
        #pragma unroll
        for (int r = 0; r < 8; ++r)
            part += __expf(rho[i0 + r + 8 * h]) * accj[r] * acck[r];
        atomicAdd(out, -part);   // result is z2 - z1
    }
}

// ---------------------------------------------------------------------------
// z_pdist2: one wave per edge (grid-stride). Each lane holds 4 of the 128
// features (float4 -> 512B coalesced row reads, all L2-resident). The next
// edge's indices are fetched before the current edge's reduction so the
// gather chain of iteration e+1 overlaps the math of iteration e.
// ---------------------------------------------------------------------------
__global__ __launch_bounds__(256)
void ldm_z2_kernel(const float* __restrict__ L, const float* __restrict__ R,
                   const float* __restrict__ U, const float* __restrict__ rho,
                   const float* __restrict__ nu, const float* __restrict__ tau,
                   const float* __restrict__ w, const int* __restrict__ si,
                   const int* __restrict__ sj, const int* __restrict__ sk,
                   int E, float* __restrict__ out)
{
    const int lane   = threadIdx.x & 31;
    const int wave   = (blockIdx.x * blockDim.x + threadIdx.x) >> 5;
    const int nwaves = (gridDim.x * blockDim.x) >> 5;

    float local = 0.0f;
    if (wave < E) {
        int i = si[wave], j = sj[wave], k = sk[wave];
        for (int e = wave; e < E; e += nwaves) {
            float4 lv = ((const float4*)(L + (size_t)i * 128))[lane];
            float4 rv = ((const float4*)(R + (size_t)j * 128))[lane];
            float4 uv = ((const float4*)(U + (size_t)k * 128))[lane];
            const int ci = i, cj = j, ck = k;

            const int e2 = e + nwaves;        // prefetch next indices
            if (e2 < E) { i = si[e2]; j = sj[e2]; k = sk[e2]; }

            float dx = lv.x - rv.x + EPS, dy = lv.y - rv.y + EPS;
            float dz = lv.z - rv.z + EPS, dw = lv.w - rv.w + EPS;
            float slr = dx*dx + dy*dy + dz*dz + dw*dw;
            dx = lv.x - uv.x + EPS; dy = lv.y - uv.y + EPS;
            dz = lv.z - uv.z + EPS; dw = lv.w - uv.w + EPS;
            float slu = dx*dx + dy*dy + dz*dz + dw*dw;

            #pragma unroll
            for (int off = 16; off; off >>= 1) {
                slr += __shfl_xor(slr, off, 32);
                slu += __shfl_xor(slu, off, 32);
            }
            if (lane == 0)
                local += w[e] * (rho[ci] + nu[cj] + tau[ck] + sqrtf(slr) + sqrtf(slu));
        }
    }
    if (lane == 0) atomicAdd(out, local);
}

extern "C" void kernel_launch(void* const* d_in, const int* in_sizes, int n_in,
                              void* d_out, int out_size, void* d_ws, size_t ws_size,
                              hipStream_t stream) {
    const float* L   = (const float*)d_in[0];
    const float* R   = (const float*)d_in[1];
    const float* U   = (const float*)d_in[2];
    const float* rho = (const float*)d_in[3];
    const float* nu  = (const float*)d_in[4];
    const float* tau = (const float*)d_in[5];
    const float* w   = (const float*)d_in[6];
    const int*   si  = (const int*)d_in[7];
    const int*   sj  = (const int*)d_in[8];
    const int*   sk  = (const int*)d_in[9];
    const int N = in_sizes[3];
    const int S = in_sizes[4];
    const int B = in_sizes[5];
    const int E = in_sizes[6];
    float* out = (float*)d_out;

    ldm_init_out<<<1, 32, 0, stream>>>(out);

    const int nib    = (N + 15) / 16;       // 1250 i-blocks
    const int blocks = (nib + 3) / 4;       // 4 waves (i-blocks) per 128-thread block
    ldm_z1_kernel<<<blocks, 128, 0, stream>>>(L, R, U, rho, nu, tau, N, S, B, out);

    ldm_z2_kernel<<<512, 256, 0, stream>>>(L, R, U, rho, nu, tau, w, si, sj, sk, E, out);
}